// vPFAE_TAG_68539088110351
// MI455X (gfx1250) — compile-verified
//
#include <hip/hip_runtime.h>

typedef __attribute__((ext_vector_type(2))) float v2f;
typedef __attribute__((ext_vector_type(8))) float v8f;

#define IN_C 96
#define L1C  70
#define L1P  72   // padded width of layer-1 activations (even, mult of 4)
#define L2C  43
#define L2P  44   // padded width of layer-2 activations
#define OUTC 16

// ---------------- elementwise / setup kernels ----------------

__global__ void k_zero(float* __restrict__ p, int n) {
  int i = blockIdx.x * blockDim.x + threadIdx.x;
  if (i < n) p[i] = 0.0f;
}

__global__ void k_deg(const int* __restrict__ col, float* __restrict__ deg, int E) {
  int i = blockIdx.x * blockDim.x + threadIdx.x;
  if (i < E) atomicAdd(&deg[col[i]], 1.0f);
}

__global__ void k_node_norm(const float* __restrict__ deg, float* __restrict__ dis,
                            float* __restrict__ dis2, int n) {
  int i = blockIdx.x * blockDim.x + threadIdx.x;
  if (i >= n) return;
  float d = deg[i];
  dis[i]  = (d > 0.0f) ? rsqrtf(fmaxf(d, 1.0f)) : 0.0f;  // gcn_norm, no self loops
  dis2[i] = rsqrtf(d + 1.0f);                            // gcn_norm with self loops
}

__global__ void k_edge_norm(const int* __restrict__ row, const int* __restrict__ col,
                            const float* __restrict__ dis, const float* __restrict__ dis2,
                            float* __restrict__ nrm, float* __restrict__ nrm2, int E) {
  int i = blockIdx.x * blockDim.x + threadIdx.x;
  if (i >= E) return;
  int r = row[i], c = col[i];
  nrm[i]  = dis[r]  * dis[c];
  nrm2[i] = dis2[r] * dis2[c];
}

// hout[col[e]][c] += w[e] * hin[row[e]][c], row stride `ld` (pad columns untouched).
// Consecutive threads cover consecutive channels of one edge -> coalesced gather.
__global__ void k_spmm(const int* __restrict__ row, const int* __restrict__ col,
                       const float* __restrict__ w, const float* __restrict__ hin,
                       float* __restrict__ hout, int E, int C, int ld) {
  int tid = blockIdx.x * blockDim.x + threadIdx.x;
  if (tid >= E * C) return;
  int e = tid / C;
  int c = tid - e * C;
  float v = w[e] * hin[(size_t)row[e] * ld + c];
  atomicAdd(&hout[(size_t)col[e] * ld + c], v);
}

__global__ void k_bias_relu(float* __restrict__ y, const float* __restrict__ b,
                            int n, int C, int ld) {
  int tid = blockIdx.x * blockDim.x + threadIdx.x;
  if (tid >= n * C) return;
  int i = tid / C, c = tid - i * C;
  float* p = y + (size_t)i * ld + c;
  *p = fmaxf(*p + b[c], 0.0f);
}

__global__ void k_gcn_final(const float* __restrict__ agg, const float* __restrict__ y,
                            const float* __restrict__ dis2, const float* __restrict__ b,
                            float* __restrict__ out, int n, int C) {
  int tid = blockIdx.x * blockDim.x + threadIdx.x;
  if (tid >= n * C) return;
  int i = tid / C, c = tid - i * C;
  float s = dis2[i];
  out[tid] = agg[tid] + s * s * y[tid] + b[c];
}

// ---------------- fp32 WMMA GEMM ----------------
// Y[M x Cout] (+)= X[M x Cin] @ W[Cin x Cout].  X has row stride ldx (even,
// >= Kpad, zero-padded), Y has row stride ldy.  W is staged into LDS zero-
// padded to Kpad x (NT*16), pair-interleaved so a lane's B fragment
// {W[k][n], W[k+1][n]} is one aligned ds b64 load.  One wave owns a 16-row
// stripe across all NT column tiles (A reused across tiles); the K loop is a
// 2-stage software pipeline so LDS/global latency overlaps the WMMAs.
// A rows are only address-clamped (never value-masked): a WMMA A-row affects
// only the same D row, and out-of-range rows are dropped by the store guard.
template <int NT>
__global__ __launch_bounds__(256)
void k_gemm_wmma(const float* __restrict__ X, const float* __restrict__ W,
                 float* __restrict__ Y, int M, int Cin, int Cout,
                 int ldx, int ldy, int accumulate) {
  extern __shared__ float lds_w[];           // Kpad * NT*16 floats
  const int Kpad  = (Cin + 3) & ~3;
  const int Coutp = NT * 16;

  // cooperative, padded, pair-interleaved weight stage:
  // lds_w[((k>>1)*Coutp + n)*2 + (k&1)] = W[k][n] (0 beyond Cin/Cout)
  for (int t = threadIdx.x; t < Kpad * Coutp; t += 256) {
    int half = t & 1;
    int pn   = t >> 1;                 // pair*Coutp + n
    int pair = pn / Coutp;
    int n    = pn - pair * Coutp;
    int k    = pair * 2 + half;
    lds_w[t] = (k < Cin && n < Cout) ? W[(size_t)k * Cout + n] : 0.0f;
  }
  __syncthreads();

  const int lane  = threadIdx.x & 31;
  const int wave  = threadIdx.x >> 5;
  const int tileM = blockIdx.x * 8 + wave;
  const int Mtiles = (M + 15) >> 4;
  if (tileM >= Mtiles) return;               // wave-uniform exit (after barrier)

  const int hi  = lane >> 4;                 // half-wave: K pairs {0,1} vs {2,3}
  const int l16 = lane & 15;
  const int mA  = tileM * 16 + l16;          // A row held by this lane
  const int mAs = (mA < M) ? mA : (M - 1);   // clamp address only

  // per-lane base pointers; step over K by adding kk (A) / (kk>>1)*Coutp*2 (B)
  const float* __restrict__ arow = X + (size_t)mAs * ldx + hi * 2;
  const float* __restrict__ brow = lds_w + ((size_t)hi * Coutp + l16) * 2;

  v8f acc[NT];
  #pragma unroll
  for (int t = 0; t < NT; ++t) {
    if (accumulate) {
      #pragma unroll
      for (int v = 0; v < 8; ++v) {
        int m = tileM * 16 + v + hi * 8;
        int n = t * 16 + l16;
        bool ok = (m < M) && (n < Cout);
        float val = Y[(size_t)(m < M ? m : M - 1) * ldy + (n < Cout ? n : Cout - 1)];
        acc[t][v] = ok ? val : 0.0f;
      }
    } else {
      #pragma unroll
      for (int v = 0; v < 8; ++v) acc[t][v] = 0.0f;
    }
  }

  // ---- 2-stage software-pipelined K loop ----
  v2f a_cur = *(const v2f*)(arow);           // kk = 0
  v2f b_cur[NT];
  #pragma unroll
  for (int t = 0; t < NT; ++t) b_cur[t] = *(const v2f*)(brow + t * 32);

  for (int kk = 0; kk + 4 < Kpad; kk += 4) {
    // prefetch next K-step while current WMMAs run
    v2f a_nxt = *(const v2f*)(arow + kk + 4);
    const float* bnxt = brow + (size_t)((kk + 4) >> 1) * Coutp * 2;
    v2f b_nxt[NT];
    #pragma unroll
    for (int t = 0; t < NT; ++t) b_nxt[t] = *(const v2f*)(bnxt + t * 32);

    #pragma unroll
    for (int t = 0; t < NT; ++t)
      acc[t] = __builtin_amdgcn_wmma_f32_16x16x4_f32(
          false, a_cur, false, b_cur[t], (short)0, acc[t], false, false);

    a_cur = a_nxt;
    #pragma unroll
    for (int t = 0; t < NT; ++t) b_cur[t] = b_nxt[t];
  }
  #pragma unroll
  for (int t = 0; t < NT; ++t)               // epilogue K-step
    acc[t] = __builtin_amdgcn_wmma_f32_16x16x4_f32(
        false, a_cur, false, b_cur[t], (short)0, acc[t], false, false);

  #pragma unroll
  for (int t = 0; t < NT; ++t) {
    #pragma unroll
    for (int v = 0; v < 8; ++v) {
      int m = tileM * 16 + v + hi * 8;
      int n = t * 16 + l16;
      if (m < M && n < Cout) Y[(size_t)m * ldy + n] = acc[t][v];
    }
  }
}

// ---------------- host-side orchestration ----------------

static inline int cdiv(int a, int b) { return (a + b - 1) / b; }

extern "C" void kernel_launch(void* const* d_in, const int* in_sizes, int n_in,
                              void* d_out, int out_size, void* d_ws, size_t ws_size,
                              hipStream_t stream) {
  const float* x   = (const float*)d_in[0];
  const int*   ei  = (const int*)  d_in[1];
  const float* W1  = (const float*)d_in[2];
  const float* b1  = (const float*)d_in[3];
  const float* W2  = (const float*)d_in[4];
  const float* b2  = (const float*)d_in[5];
  const float* Wmu = (const float*)d_in[6];
  const float* bmu = (const float*)d_in[7];
  const float* Wls = (const float*)d_in[8];
  const float* bls = (const float*)d_in[9];

  const int N = in_sizes[0] / IN_C;
  const int E = in_sizes[1] / 2;
  const int* row = ei;
  const int* col = ei + E;

  // workspace carving (fp32)
  float* ws = (float*)d_ws;
  size_t o = 0;
  float* deg  = ws + o; o += (size_t)N;
  float* dis  = ws + o; o += (size_t)N;
  float* dis2 = ws + o; o += (size_t)N;
  float* nrm  = ws + o; o += (size_t)E;
  float* nrm2 = ws + o; o += (size_t)E;
  float* hA   = ws + o; o += (size_t)N * IN_C;   // propagation ping  (stride 96 / 72)
  float* hB   = ws + o; o += (size_t)N * IN_C;   // propagation pong
  float* h1   = ws + o; o += (size_t)N * L1P;    // layer-1 out, padded stride 72
  float* h2   = ws + o; o += (size_t)N * L2P;    // layer-2 out, padded stride 44
  float* ymu  = ws + o; o += (size_t)N * OUTC;
  float* yls  = ws + o; o += (size_t)N * OUTC;
  float* agg  = ws + o; o += (size_t)N * OUTC;

  const int TB = 256;
  const dim3 gemm_grid(cdiv(cdiv(N, 16), 8));
  const int sh1 = IN_C * 5 * 16 * 4;             // 96*80 floats = 30 KB
  const int sh2 = L1P  * 3 * 16 * 4;             // 72*48 floats = 13.5 KB
  const int shh = L2P  * 1 * 16 * 4;             // 44*16 floats = 2.75 KB
  #define ZERO(p, n) k_zero<<<cdiv((n), TB), TB, 0, stream>>>((p), (n))
  #define SPMM(w_, in_, out_, C_, LD_) \
    k_spmm<<<cdiv((E) * (C_), TB), TB, 0, stream>>>(row, col, (w_), (in_), (out_), E, (C_), (LD_))

  // degrees + normalizations (shared by every propagation)
  ZERO(deg, N);
  k_deg<<<cdiv(E, TB), TB, 0, stream>>>(col, deg, E);
  k_node_norm<<<cdiv(N, TB), TB, 0, stream>>>(deg, dis, dis2, N);
  k_edge_norm<<<cdiv(E, TB), TB, 0, stream>>>(row, col, dis, dis2, nrm, nrm2, E);

  // zero padded activation buffers so pad columns are 0 for the next GEMM's A reads
  ZERO(h1, N * L1P);
  ZERO(h2, N * L2P);

  // ---- TAGConv layer 1: 96 -> 70, K=3 hops (prop width 96, stride 96)
  k_gemm_wmma<5><<<gemm_grid, 256, sh1, stream>>>(x,  W1 + 0 * IN_C * L1C, h1, N, IN_C, L1C, IN_C, L1P, 0);
  ZERO(hA, N * IN_C); SPMM(nrm, x,  hA, IN_C, IN_C);
  k_gemm_wmma<5><<<gemm_grid, 256, sh1, stream>>>(hA, W1 + 1 * IN_C * L1C, h1, N, IN_C, L1C, IN_C, L1P, 1);
  ZERO(hB, N * IN_C); SPMM(nrm, hA, hB, IN_C, IN_C);
  k_gemm_wmma<5><<<gemm_grid, 256, sh1, stream>>>(hB, W1 + 2 * IN_C * L1C, h1, N, IN_C, L1C, IN_C, L1P, 1);
  ZERO(hA, N * IN_C); SPMM(nrm, hB, hA, IN_C, IN_C);
  k_gemm_wmma<5><<<gemm_grid, 256, sh1, stream>>>(hA, W1 + 3 * IN_C * L1C, h1, N, IN_C, L1C, IN_C, L1P, 1);
  k_bias_relu<<<cdiv(N * L1C, TB), TB, 0, stream>>>(h1, b1, N, L1C, L1P);

  // ---- TAGConv layer 2: 70 -> 43, K=3 hops (prop width 70, stride 72, pad stays 0)
  k_gemm_wmma<3><<<gemm_grid, 256, sh2, stream>>>(h1, W2 + 0 * L1C * L2C, h2, N, L1C, L2C, L1P, L2P, 0);
  ZERO(hA, N * L1P); SPMM(nrm, h1, hA, L1C, L1P);
  k_gemm_wmma<3><<<gemm_grid, 256, sh2, stream>>>(hA, W2 + 1 * L1C * L2C, h2, N, L1C, L2C, L1P, L2P, 1);
  ZERO(hB, N * L1P); SPMM(nrm, hA, hB, L1C, L1P);
  k_gemm_wmma<3><<<gemm_grid, 256, sh2, stream>>>(hB, W2 + 2 * L1C * L2C, h2, N, L1C, L2C, L1P, L2P, 1);
  ZERO(hA, N * L1P); SPMM(nrm, hB, hA, L1C, L1P);
  k_gemm_wmma<3><<<gemm_grid, 256, sh2, stream>>>(hA, W2 + 3 * L1C * L2C, h2, N, L1C, L2C, L1P, L2P, 1);
  k_bias_relu<<<cdiv(N * L2C, TB), TB, 0, stream>>>(h2, b2, N, L2C, L2P);

  // ---- GCN head: mu
  float* mu_out = (float*)d_out;
  float* ls_out = (float*)d_out + (size_t)N * OUTC;
  k_gemm_wmma<1><<<gemm_grid, 256, shh, stream>>>(h2, Wmu, ymu, N, L2C, OUTC, L2P, OUTC, 0);
  ZERO(agg, N * OUTC); SPMM(nrm2, ymu, agg, OUTC, OUTC);
  k_gcn_final<<<cdiv(N * OUTC, TB), TB, 0, stream>>>(agg, ymu, dis2, bmu, mu_out, N, OUTC);

  // ---- GCN head: logstd
  k_gemm_wmma<1><<<gemm_grid, 256, shh, stream>>>(h2, Wls, yls, N, L2C, OUTC, L2P, OUTC, 0);
  ZERO(agg, N * OUTC); SPMM(nrm2, yls, agg, OUTC, OUTC);
  k_gcn_final<<<cdiv(N * OUTC, TB), TB, 0, stream>>>(agg, yls, dis2, bls, ls_out, N, OUTC);

  #undef ZERO
  #undef SPMM
}